// VectorQuantizerEMA_72447508349367
// MI455X (gfx1250) — compile-verified
//
#include <hip/hip_runtime.h>
#include <hip/hip_bf16.h>
#include <math.h>

typedef float v2f __attribute__((ext_vector_type(2)));
typedef float v8f __attribute__((ext_vector_type(8)));

#define KC 1024
#define DD 256
#define NN 131072        // B*T = 64*2048
#define DECAYF 0.99f
#define OMDF   0.01f
#define EPSF   1e-5f
#define COMMITF 0.25f

// ---- output layout (floats, concatenated in reference return order) ----
constexpr size_t OFF_VQ   = 0;                               // vq_loss (1)
constexpr size_t OFF_Q    = 1;                               // quantized_out [D,B,T]
constexpr size_t SZ_Q     = (size_t)DD * NN;
constexpr size_t OFF_PERP = OFF_Q + SZ_Q;                    // perplexity (1)
constexpr size_t OFF_ENC  = OFF_PERP + 1;                    // encodings [N,K]
constexpr size_t SZ_NK    = (size_t)NN * KC;
constexpr size_t OFF_DIST = OFF_ENC + SZ_NK;                 // distances [N,K]
constexpr size_t OFF_IDX  = OFF_DIST + SZ_NK;                // encoding_indices [N]
constexpr size_t OFF_NCS  = OFF_IDX + NN;                    // new_cs [K]
constexpr size_t OFF_NEW  = OFF_NCS + KC;                    // new_ema_w [K,D]
constexpr size_t OFF_NEMB = OFF_NEW + (size_t)KC * DD;       // new_embedding [K,D]

// ---- dist kernel geometry ----
#define WGT  128          // 4 wave32s
#define ROWS 64           // rows (encoder vectors) per workgroup
#define AST  258          // padded LDS row stride (even -> 8B-aligned b64 frags, ~2-way conflicts)

// =====================================================================
// workspace zeroing (counts, dw, loss accumulator) — rerun every call
// =====================================================================
__global__ void vq_zero_kernel(float* __restrict__ ws_counts,
                               float* __restrict__ ws_dw,
                               float* __restrict__ ws_scalar) {
    size_t i = (size_t)blockIdx.x * blockDim.x + threadIdx.x;
    if (i < (size_t)KC * DD) ws_dw[i] = 0.f;
    if (i < KC)              ws_counts[i] = 0.f;
    if (i < 16)              ws_scalar[i] = 0.f;
}

// =====================================================================
// per-code squared norms ||e_k||^2
// =====================================================================
__global__ void vq_esq_kernel(const float* __restrict__ emb,
                              float* __restrict__ esq) {
    int k = blockIdx.x * blockDim.x + threadIdx.x;
    if (k < KC) {
        const float* e = emb + (size_t)k * DD;
        float s = 0.f;
        for (int d = 0; d < DD; ++d) s += e[d] * e[d];
        esq[k] = s;
    }
}

// =====================================================================
// main kernel: fp32 WMMA distance GEMM + argmin + distances/encodings
// streams + counts/dw scatter
// =====================================================================
__global__ void __launch_bounds__(WGT)
vq_dist_kernel(const float* __restrict__ inp,     // [D, N]   (D,B,T flat)
               const float* __restrict__ emb,     // [K, D]
               const float* __restrict__ esq_g,   // [K]
               float* __restrict__ out,
               int*   __restrict__ ws_idx,
               float* __restrict__ ws_counts,
               float* __restrict__ ws_dw)
{
    extern __shared__ float lds[];
    float* Ash  = lds;                       // 64 x 258
    float* xsq  = Ash + ROWS * AST;          // 64
    float* Bsh  = xsq + ROWS;                // 16 x 258
    float* esqs = Bsh + 16 * AST;            // 1024
    int*   winner = (int*)(esqs + KC);       // 64

    const int tid  = threadIdx.x;
    const int lane = tid & 31;
    const int wave = tid >> 5;               // 0..3
    const int l16  = lane & 15;
    const int lh   = lane >> 4;              // lane half (K split / M split)
    const int n0   = blockIdx.x * ROWS;
    const int rowbase = wave * 16;           // 16 rows per wave

    // ---- stage A tile: flat[n][d] = inputs[d*N + n] ----
    for (int e = tid; e < ROWS * DD; e += WGT) {
        int d = e >> 6, i = e & 63;
        Ash[i * AST + d] = inp[(size_t)d * NN + n0 + i];
    }
    for (int e = tid; e < KC; e += WGT) esqs[e] = esq_g[e];
    __syncthreads();

    if (tid < ROWS) {
        float s = 0.f;
        for (int d = 0; d < DD; ++d) { float v = Ash[tid * AST + d]; s += v * v; }
        xsq[tid] = s;
    }
    __syncthreads();

    float minv[8]; int mini[8];
#pragma unroll
    for (int r = 0; r < 8; ++r) { minv[r] = 3.4e38f; mini[r] = 0; }

    float* out_dist = out + OFF_DIST;

    for (int ct = 0; ct < KC / 16; ++ct) {
        const int cbase = ct * 16;
        // cooperative B tile load: 16 codes x 256 dims
        for (int e = tid; e < 16 * DD; e += WGT) {
            int r = e >> 8, d = e & 255;
            Bsh[r * AST + d] = emb[(size_t)(cbase + r) * DD + d];
        }
        __syncthreads();
        if (ct + 1 < KC / 16 && tid < 16)
            __builtin_prefetch(&emb[(size_t)(cbase + 16 + tid) * DD], 0, 3);

        // ---- 16x16 output tile per wave, K-loop over D in steps of 4 ----
        v8f acc = {};
        const float* Ap = Ash + (rowbase + l16) * AST + lh * 2; // A: M=lane%16, K=lh*2+v
        const float* Bp = Bsh + l16 * AST + lh * 2;             // B: N=lane%16, K=lh*2+v
#pragma unroll 8
        for (int kk = 0; kk < DD; kk += 4) {
            v2f a = *(const v2f*)(Ap + kk);
            v2f b = *(const v2f*)(Bp + kk);
            acc = __builtin_amdgcn_wmma_f32_16x16x4_f32(
                false, a, false, b, (short)0, acc, false, false);
        }

        const float ec = esqs[cbase + l16];
        const int   c  = cbase + l16;
#pragma unroll
        for (int r = 0; r < 8; ++r) {
            int lrow = rowbase + r + 8 * lh;     // C layout: M = r + 8*(lane>=16)
            float dist = xsq[lrow] + ec - 2.0f * acc[r];
            __builtin_nontemporal_store(dist, &out_dist[(size_t)(n0 + lrow) * KC + c]);
            if (dist < minv[r] || (dist == minv[r] && c < mini[r])) {
                minv[r] = dist; mini[r] = c;
            }
        }
        __syncthreads();   // protect Bsh before next overwrite
    }

    // ---- argmin across the 16 lanes holding one row's columns ----
#pragma unroll
    for (int r = 0; r < 8; ++r) {
#pragma unroll
        for (int m = 1; m < 16; m <<= 1) {
            float ov = __shfl_xor(minv[r], m, 32);
            int   oi = __shfl_xor(mini[r], m, 32);
            if (ov < minv[r] || (ov == minv[r] && oi < mini[r])) {
                minv[r] = ov; mini[r] = oi;
            }
        }
    }
    if (l16 == 0) {
#pragma unroll
        for (int r = 0; r < 8; ++r) winner[rowbase + r + 8 * lh] = mini[r];
    }
    __syncthreads();

    // indices (as float output + int ws) and histogram counts
    if (tid < ROWS) {
        int wi = winner[tid];
        out[OFF_IDX + n0 + tid] = (float)wi;
        ws_idx[n0 + tid] = wi;
        atomicAdd(&ws_counts[wi], 1.0f);
    }
    // one-hot encodings stream (536 MB total -> nontemporal)
    float* out_enc = out + OFF_ENC;
    for (int e = tid; e < ROWS * KC; e += WGT) {
        int i = e >> 10, j = e & 1023;
        float v = (j == winner[i]) ? 1.0f : 0.0f;
        __builtin_nontemporal_store(v, &out_enc[(size_t)(n0 + i) * KC + j]);
    }
    // dw scatter-add: dw[winner[i]] += flat[i]
    for (int e = tid; e < ROWS * DD; e += WGT) {
        int i = e >> 8, d = e & 255;
        atomicAdd(&ws_dw[(size_t)winner[i] * DD + d], Ash[i * AST + d]);
    }
}

// =====================================================================
// EMA cluster size + Laplace smoothing + perplexity (single WG, K=1024)
// =====================================================================
__global__ void __launch_bounds__(1024)
vq_ema_cs_kernel(const float* __restrict__ ema_cs,
                 const float* __restrict__ ws_counts,
                 float* __restrict__ out,
                 float* __restrict__ ws_newcs)
{
    __shared__ float red[1024];
    int t = threadIdx.x;
    float cnt = ws_counts[t];
    float c = ema_cs[t] * DECAYF + OMDF * cnt;
    red[t] = c;
    __syncthreads();
    for (int s = 512; s > 0; s >>= 1) { if (t < s) red[t] += red[t + s]; __syncthreads(); }
    float nsum = red[0];
    __syncthreads();
    float ncs = (c + EPSF) / (nsum + KC * EPSF) * nsum;
    out[OFF_NCS + t] = ncs;
    ws_newcs[t] = ncs;

    float p = cnt / (float)NN;
    red[t] = p * logf(p + 1e-10f);
    __syncthreads();
    for (int s = 512; s > 0; s >>= 1) { if (t < s) red[t] += red[t + s]; __syncthreads(); }
    if (t == 0) out[OFF_PERP] = expf(-red[0]);
}

// =====================================================================
// new_ema_w and new_embedding (K*D elements)
// =====================================================================
__global__ void vq_emaw_kernel(const float* __restrict__ ema_w,
                               const float* __restrict__ ws_dw,
                               const float* __restrict__ ws_newcs,
                               float* __restrict__ out)
{
    size_t e = (size_t)blockIdx.x * blockDim.x + threadIdx.x;
    if (e >= (size_t)KC * DD) return;
    int k = (int)(e >> 8);
    float w = ema_w[e] * DECAYF + OMDF * ws_dw[e];
    out[OFF_NEW  + e] = w;
    out[OFF_NEMB + e] = w / ws_newcs[k];
}

// =====================================================================
// quantize with updated codebook + commitment-loss accumulation
// =====================================================================
__global__ void __launch_bounds__(256)
vq_quant_kernel(const float* __restrict__ inp,
                const int*   __restrict__ ws_idx,
                float* __restrict__ out,
                float* __restrict__ ws_scalar)
{
    __shared__ float red[256];
    int n = blockIdx.x * 256 + threadIdx.x;
    const float* eb = out + OFF_NEMB + (size_t)ws_idx[n] * DD;   // L2-resident 1MB table
    float* outq = out + OFF_Q;
    float accv = 0.f;
    for (int d = 0; d < DD; ++d) {
        float q = eb[d];
        float x = inp[(size_t)d * NN + n];
        outq[(size_t)d * NN + n] = q;       // straight-through value == quantized
        float df = q - x;
        accv += df * df;
    }
    red[threadIdx.x] = accv;
    __syncthreads();
    for (int s = 128; s > 0; s >>= 1) {
        if (threadIdx.x < s) red[threadIdx.x] += red[threadIdx.x + s];
        __syncthreads();
    }
    if (threadIdx.x == 0) atomicAdd(ws_scalar, red[0]);
}

__global__ void vq_final_kernel(const float* __restrict__ ws_scalar,
                                float* __restrict__ out) {
    out[OFF_VQ] = COMMITF * ws_scalar[0] / ((float)NN * (float)DD);
}

// =====================================================================
extern "C" void kernel_launch(void* const* d_in, const int* in_sizes, int n_in,
                              void* d_out, int out_size, void* d_ws, size_t ws_size,
                              hipStream_t stream)
{
    const float* inp    = (const float*)d_in[0];   // [D,B,T]
    const float* emb    = (const float*)d_in[1];   // [K,D]
    const float* ema_cs = (const float*)d_in[2];   // [K]
    const float* ema_w  = (const float*)d_in[3];   // [K,D]
    float* out = (float*)d_out;

    float* ws        = (float*)d_ws;
    float* ws_counts = ws;                         // K
    float* ws_esq    = ws + 1024;                  // K
    float* ws_newcs  = ws + 2048;                  // K
    float* ws_scalar = ws + 3072;                  // 16
    int*   ws_idx    = (int*)(ws + 3088);          // N ints
    float* ws_dw     = ws + 3088 + NN;             // K*D

    vq_zero_kernel<<<(KC * DD + 255) / 256, 256, 0, stream>>>(ws_counts, ws_dw, ws_scalar);
    vq_esq_kernel<<<(KC + 255) / 256, 256, 0, stream>>>(emb, ws_esq);

    size_t shbytes = (size_t)(ROWS * AST + ROWS + 16 * AST + KC + 64) * sizeof(float);
    vq_dist_kernel<<<NN / ROWS, WGT, shbytes, stream>>>(inp, emb, ws_esq, out,
                                                        ws_idx, ws_counts, ws_dw);

    vq_ema_cs_kernel<<<1, 1024, 0, stream>>>(ema_cs, ws_counts, out, ws_newcs);
    vq_emaw_kernel<<<(KC * DD) / 256, 256, 0, stream>>>(ema_w, ws_dw, ws_newcs, out);
    vq_quant_kernel<<<NN / 256, 256, 0, stream>>>(inp, ws_idx, out, ws_scalar);
    vq_final_kernel<<<1, 1, 0, stream>>>(ws_scalar, out);
}